// MSE_evs_69329362092103
// MI455X (gfx1250) — compile-verified
//
#include <hip/hip_runtime.h>
#include <hip/hip_bf16.h>
#include <math.h>

// Problem constants from the reference:
//   B=8, F=8192, N=16, EVS=256 selected matrices per batch -> 2048 matrices.
#define BATCH 8
#define FDIM  8192
#define NDIM  16
#define EVS   256
#define NMAT  (BATCH * EVS)          // 2048
#define TOTAL_EVS (NMAT * NDIM)      // 32768 elements averaged

typedef __attribute__((ext_vector_type(2))) float v2f;
typedef __attribute__((ext_vector_type(8))) float v8f;

// One wave32 per 16x16 complex matrix. Shifted complex QR with deflation.
// Similarity update A' = R * G^H done with V_WMMA_F32_16X16X4_F32 chains.
__global__ __launch_bounds__(32) void eig16_qr_wmma_kernel(
    const float* __restrict__ y_re, const float* __restrict__ y_im,
    const float* __restrict__ y_true, const int* __restrict__ idxs,
    float* __restrict__ partial)
{
    __shared__ float Ar[16][17], Ai[16][17];   // matrix (becomes R in-place)
    __shared__ float Gr[16][17], Gi[16][17];   // accumulated rotations

    const int w    = blockIdx.x;       // matrix id 0..2047
    const int b    = w >> 8;           // batch
    const int j    = w & 255;          // selected-ev slot
    const int lane = threadIdx.x;      // 0..31
    const int col  = lane & 15;
    const bool hi  = lane >= 16;

    const int f = idxs[j];
    const size_t base = ((size_t)b * FDIM + (size_t)f) * (NDIM * NDIM);

    // ---- load 16x16 complex matrix: 8 floats/lane re + im (float4 pairs) ----
    {
        const int e0 = lane * 8;
        const float4* pr = (const float4*)(y_re + base + e0);
        const float4* pi = (const float4*)(y_im + base + e0);
        float4 r0 = pr[0], r1 = pr[1];
        float4 i0 = pi[0], i1 = pi[1];
        float vr[8] = {r0.x, r0.y, r0.z, r0.w, r1.x, r1.y, r1.z, r1.w};
        float vi[8] = {i0.x, i0.y, i0.z, i0.w, i1.x, i1.y, i1.z, i1.w};
        #pragma unroll
        for (int t = 0; t < 8; ++t) {
            int e = e0 + t;
            Ar[e >> 4][e & 15] = vr[t];
            Ai[e >> 4][e & 15] = vi[t];
        }
    }
    __syncthreads();

    // ---- shifted QR iteration with bottom deflation ----
    int m = NDIM;
    int iter = 0;
    while (m > 1 && iter < 240) {
        // deflation test on subdiagonal A[m-1][m-2]
        float sr = Ar[m-1][m-2], si = Ai[m-1][m-2];
        float sub = sqrtf(sr*sr + si*si);
        float daR = Ar[m-2][m-2], daI = Ai[m-2][m-2];
        float ddR = Ar[m-1][m-1], ddI = Ai[m-1][m-1];
        float d1 = sqrtf(daR*daR + daI*daI);
        float d2 = sqrtf(ddR*ddR + ddI*ddI);
        if (sub <= 4e-7f * (d1 + d2) + 1e-34f) { m--; continue; }
        iter++;

        // Wilkinson shift from trailing 2x2 [[a,b],[c,d]] of active block
        float ar_ = daR, ai_ = daI;
        float br_ = Ar[m-2][m-1], bi_ = Ai[m-2][m-1];
        float cr_ = sr,  ci_ = si;
        float dr_ = ddR, di_ = ddI;
        float der = 0.5f*(ar_ - dr_), dei = 0.5f*(ai_ - di_);
        float bcr = br_*cr_ - bi_*ci_, bci = br_*ci_ + bi_*cr_;
        float zr = der*der - dei*dei + bcr;
        float zi = 2.0f*der*dei + bci;
        float zm = sqrtf(zr*zr + zi*zi);
        float wr = sqrtf(fmaxf(0.5f*(zm + zr), 0.0f));
        float wi = (zi >= 0.0f ? 1.0f : -1.0f) * sqrtf(fmaxf(0.5f*(zm - zr), 0.0f));
        float p1 = (der+wr)*(der+wr) + (dei+wi)*(dei+wi);
        float p2 = (der-wr)*(der-wr) + (dei-wi)*(dei-wi);
        float midr = 0.5f*(ar_ + dr_), midi = 0.5f*(ai_ + di_);
        float sgr, sgi;
        if (p1 < p2) { sgr = midr + wr; sgi = midi + wi; }
        else         { sgr = midr - wr; sgi = midi - wi; }

        // A -= sigma*I on active diagonal; G = I
        if (lane < m) { Ar[lane][lane] -= sgr; Ai[lane][lane] -= sgi; }
        #pragma unroll
        for (int t = 0; t < 8; ++t) {
            int e = lane * 8 + t;
            int rr = e >> 4, cc = e & 15;
            Gr[rr][cc] = (rr == cc) ? 1.0f : 0.0f;
            Gi[rr][cc] = 0.0f;
        }
        __syncthreads();

        // ---- complex Givens QR of the active m x m block: G*(A-sI) = R ----
        for (int k = 0; k < m - 1; ++k) {
            for (int i = m - 1; i > k; --i) {
                float xr = Ar[i-1][k], xi = Ai[i-1][k];
                float yr = Ar[i][k],   yi = Ai[i][k];
                float ny2 = yr*yr + yi*yi;
                if (ny2 > 1e-32f) {
                    float nx2 = xr*xr + xi*xi;
                    float rno = sqrtf(nx2 + ny2);
                    float ch, s_r, s_i;
                    if (nx2 > 1e-32f) {
                        float nx = sqrtf(nx2);
                        ch = nx / rno;
                        float inv = 1.0f / (nx * rno);
                        s_r = (xr*yr + xi*yi) * inv;   // (x*conj(y)).re / (|x| r)
                        s_i = (xi*yr - xr*yi) * inv;   // (x*conj(y)).im / (|x| r)
                    } else {
                        ch = 0.0f;
                        float invy = 1.0f / sqrtf(ny2);
                        s_r =  yr * invy;              // conj(y)/|y|
                        s_i = -yi * invy;
                    }
                    // rotate rows (i-1, i): lanes 0-15 own row i-1, 16-31 own row i
                    float u_r = Ar[i-1][col], u_i = Ai[i-1][col];
                    float v_r = Ar[i][col],   v_i = Ai[i][col];
                    float gu_r = Gr[i-1][col], gu_i = Gi[i-1][col];
                    float gv_r = Gr[i][col],   gv_i = Gi[i][col];
                    if (!hi) {
                        Ar[i-1][col] = ch*u_r + (s_r*v_r - s_i*v_i);
                        Ai[i-1][col] = ch*u_i + (s_r*v_i + s_i*v_r);
                        Gr[i-1][col] = ch*gu_r + (s_r*gv_r - s_i*gv_i);
                        Gi[i-1][col] = ch*gu_i + (s_r*gv_i + s_i*gv_r);
                    } else {
                        // new row i = -conj(s)*u + c*v, conj(s) = (s_r, -s_i)
                        Ar[i][col] = ch*v_r - (s_r*u_r + s_i*u_i);
                        Ai[i][col] = ch*v_i - (s_r*u_i - s_i*u_r);
                        Gr[i][col] = ch*gv_r - (s_r*gu_r + s_i*gu_i);
                        Gi[i][col] = ch*gv_i - (s_r*gu_i - s_i*gu_r);
                    }
                }
            }
        }
        __syncthreads();

        // ---- A' = R * G^H via WMMA f32 16x16x4 chains ----
        // Complex: Cr = Rr*(Gr^T) + Ri*(Gi^T) ; Ci = Ri*(Gr^T) - Rr*(Gi^T)
        // (since G^H means B_re = Gr^T, B_im = -Gi^T; minus folded into loads)
        v8f crA = {};
        v8f ciA = {};
        #pragma unroll
        for (int kk = 0; kk < 16; kk += 4) {
            const int k0 = kk + (hi ? 2 : 0);
            v2f aRr, aRi, bP, bMt, bMn;
            aRr.x = Ar[col][k0]; aRr.y = Ar[col][k0+1];   // A-op: M=col, K=k0..k0+1
            aRi.x = Ai[col][k0]; aRi.y = Ai[col][k0+1];
            bP.x  = Gr[col][k0]; bP.y  = Gr[col][k0+1];   // B-op: N=col, B[k][N]=Gr[N][k]
            bMt.x = Gi[col][k0]; bMt.y = Gi[col][k0+1];
            bMn.x = -bMt.x;      bMn.y = -bMt.y;
            crA = __builtin_amdgcn_wmma_f32_16x16x4_f32(false, aRr, false, bP,  (short)0, crA, false, false);
            crA = __builtin_amdgcn_wmma_f32_16x16x4_f32(false, aRi, false, bMt, (short)0, crA, false, false);
            ciA = __builtin_amdgcn_wmma_f32_16x16x4_f32(false, aRi, false, bP,  (short)0, ciA, false, false);
            ciA = __builtin_amdgcn_wmma_f32_16x16x4_f32(false, aRr, false, bMn, (short)0, ciA, false, false);
        }
        // C/D layout: lanes 0-15 rows 0-7, lanes 16-31 rows 8-15, N = col
        const int r0 = hi ? 8 : 0;
        #pragma unroll
        for (int t = 0; t < 8; ++t) {
            Ar[r0 + t][col] = crA[t];
            Ai[r0 + t][col] = ciA[t];
        }
        __syncthreads();
        // restore shift on active diagonal
        if (lane < m) { Ar[lane][lane] += sgr; Ai[lane][lane] += sgi; }
        __syncthreads();
    }

    // ---- eigenvalue magnitudes from diagonal, squared error, wave reduce ----
    float local = 0.0f;
    if (lane < NDIM) {
        float drr = Ar[lane][lane], dii = Ai[lane][lane];
        float ev = sqrtf(drr*drr + dii*dii);
        float tv = y_true[((size_t)b * FDIM + (size_t)f) * NDIM + lane];
        float df = ev - tv;
        local = df * df;
    }
    #pragma unroll
    for (int off = 16; off >= 1; off >>= 1)
        local += __shfl_down(local, off, 32);
    if (lane == 0) partial[w] = local;
}

__global__ __launch_bounds__(256) void reduce_kernel(
    const float* __restrict__ partial, float* __restrict__ out)
{
    __shared__ float s[256];
    float v = 0.0f;
    for (int i = threadIdx.x; i < NMAT; i += 256) v += partial[i];
    s[threadIdx.x] = v;
    __syncthreads();
    for (int st = 128; st > 0; st >>= 1) {
        if ((int)threadIdx.x < st) s[threadIdx.x] += s[threadIdx.x + st];
        __syncthreads();
    }
    if (threadIdx.x == 0) out[0] = s[0] * (1.0f / (float)TOTAL_EVS);
}

extern "C" void kernel_launch(void* const* d_in, const int* in_sizes, int n_in,
                              void* d_out, int out_size, void* d_ws, size_t ws_size,
                              hipStream_t stream) {
    (void)in_sizes; (void)n_in; (void)out_size; (void)ws_size;
    const float* y_re   = (const float*)d_in[0];
    const float* y_im   = (const float*)d_in[1];
    const float* y_true = (const float*)d_in[2];
    const int*   idxs   = (const int*)d_in[3];
    float* out     = (float*)d_out;
    float* partial = (float*)d_ws;   // NMAT floats of scratch

    eig16_qr_wmma_kernel<<<NMAT, 32, 0, stream>>>(y_re, y_im, y_true, idxs, partial);
    reduce_kernel<<<1, 256, 0, stream>>>(partial, out);
}